// PhysicsConvolution_38405597561664
// MI455X (gfx1250) — compile-verified
//
#include <hip/hip_runtime.h>
#include <hip/hip_bf16.h>
#include <stdint.h>

// ---------------------------------------------------------------------------
// PhysicsConvolution for MI455X (gfx1250, wave32)
//   X0  = notes @ w                          (fp32 WMMA 16x16x4, LDS-fed)
//   X1  = segment_sum(X0[dst]*ew, src) + b   (L2-resident f32 atomics)
//   out = concat(relu(X1), X0[80000:])
//
// GEMM stage uses the CDNA5 async tensor path: the per-wave 16x128 A tile is
// copied global->LDS with GLOBAL_LOAD_ASYNC_TO_LDS_B128 (ASYNCcnt), and w is
// staged pair-interleaved so every B fragment is a single aligned ds_load_b64
// (no v_mov re-pairing in front of v_wmma).
// ---------------------------------------------------------------------------

typedef __attribute__((ext_vector_type(2))) float v2f;
typedef __attribute__((ext_vector_type(8))) float v8f;

#define N_NODES_C 100000
#define D_C       128
#define GARMENT_C 80000

#define KP       64     // number of even-k pair-rows of w (128/2)
#define BSTRIDE  144    // pair-row stride in pairs: lanes 0-15 / 16-31 -> disjoint banks
#define ASTRIDE  132    // A-tile row stride in floats: 528B (16B aligned, conflict-free)

// -------------------------------------------------------------------------
// Kernel 0: seed accumulator rows [0, N) of out with the bias vector b.
// -------------------------------------------------------------------------
__global__ __launch_bounds__(256) void pc_init_bias(const float* __restrict__ b,
                                                    float* __restrict__ out,
                                                    int n_vec4) {
    int i = blockIdx.x * 256 + threadIdx.x;          // float4 index
    if (i >= n_vec4) return;
    int col4 = i & 31;                               // 128/4 = 32 float4 per row
    ((float4*)out)[i] = ((const float4*)b)[col4];
}

// -------------------------------------------------------------------------
// Kernel 1: X0 = notes @ w via V_WMMA_F32_16X16X4_F32.
// One wave owns a 16-row M-tile across all 128 columns (8 v8f accumulators).
//
// Fragment layouts (ISA 7.12.2, wave32):
//   A 16x4:  lane L, reg v : M = L&15,  K = (L>>4)*2 + v
//   B 4x16:  lane L, reg v : N = L&15,  K = (L>>4)*2 + v
//   C 16x16: lane L, reg r : N = L&15,  M = (L>>4)*8 + r
// -------------------------------------------------------------------------
__global__ __launch_bounds__(256) void pc_gemm_wmma(const float* __restrict__ notes,
                                                    const float* __restrict__ w,
                                                    float* __restrict__ X0,
                                                    float* __restrict__ out,
                                                    int n_nodes) {
    // w pairs: (k2, n) -> { w[2*k2][n], w[2*k2+1][n] }    72 KB
    __shared__ __align__(16) float swp[KP * BSTRIDE * 2];
    // per-wave 16x128 A tiles, rows padded to ASTRIDE     66 KB
    __shared__ __align__(16) float sa[8 * 16 * ASTRIDE];

    const int tid   = threadIdx.x;
    const int wave  = tid >> 5;
    const int lane  = tid & 31;
    const int tileM = blockIdx.x * 8 + wave;         // 16-row tile index
    const bool active = (tileM * 16 < n_nodes);      // wave-uniform

    // (1) Async-stage this wave's A tile: 16 rows x 512B, one b128 per lane
    //     per row, straight into LDS (no VGPR round trip, tracked by ASYNCcnt).
    if (active) {
        const float*   gsrc  = notes + (size_t)tileM * 16 * D_C;
        const uint32_t lbase = (uint32_t)(uintptr_t)(sa + wave * 16 * ASTRIDE);
        #pragma unroll
        for (int r = 0; r < 16; ++r) {
            uint64_t ga = (uint64_t)(uintptr_t)(gsrc + r * D_C + lane * 4);
            uint32_t la = lbase + (uint32_t)(r * ASTRIDE + lane * 4) * 4u;
            asm volatile("global_load_async_to_lds_b128 %0, %1, off"
                         :: "v"(la), "v"(ga) : "memory");
        }
    }

    // (2) Stage w pair-interleaved (overlaps with the async A copies):
    //     swp[(k2*BSTRIDE + n)*2 + {0,1}] = w[2*k2][n], w[2*k2+1][n]
    #pragma unroll
    for (int j = 0; j < 32; ++j) {
        int p  = j * 256 + tid;                      // 8192 pairs total
        int k2 = p >> 7;
        int n  = p & 127;
        float2 pr;
        pr.x = w[(k2 * 2 + 0) * D_C + n];
        pr.y = w[(k2 * 2 + 1) * D_C + n];
        *(float2*)(swp + (k2 * BSTRIDE + n) * 2) = pr;
    }

    asm volatile("s_wait_asynccnt 0x0" ::: "memory"); // own async copies done
    __syncthreads();                                  // everyone's copies done
    if (!active) return;                              // EXEC stays all-1s

    const int half = lane >> 4;                      // 0 | 1
    const int l16  = lane & 15;
    const float* sArow = sa + wave * 16 * ASTRIDE + (size_t)l16 * ASTRIDE;

    v8f acc[8] = {};                                 // 16x128 fp32 accumulator

    for (int k0 = 0; k0 < D_C; k0 += 4) {
        const int ka = k0 + (half << 1);
        v2f a = *(const v2f*)(sArow + ka);           // one ds_load_b64
        const float* bb = swp + (size_t)(((k0 >> 1) + half) * BSTRIDE) * 2;
        #pragma unroll
        for (int t = 0; t < 8; ++t) {
            v2f bf = *(const v2f*)(bb + (t * 16 + l16) * 2);  // one ds_load_b64
            acc[t] = __builtin_amdgcn_wmma_f32_16x16x4_f32(
                false, a, false, bf, (short)0, acc[t], false, false);
        }
    }

    // Store C: for fixed r, lanes 0-15 write 64 contiguous bytes.
    const int baseRow = tileM * 16 + (half << 3);
    #pragma unroll
    for (int t = 0; t < 8; ++t) {
        const int col = t * 16 + l16;
        #pragma unroll
        for (int r = 0; r < 8; ++r)
            X0[(size_t)(baseRow + r) * D_C + col] = acc[t][r];
    }

    // Concat tail: X0 row r (r >= 80000) -> out row r + (N_NODES - GARMENT).
    // GARMENT is 16-aligned -> wave-uniform branch.
    if (tileM >= (GARMENT_C / 16)) {
        #pragma unroll
        for (int t = 0; t < 8; ++t) {
            const int col = t * 16 + l16;
            #pragma unroll
            for (int r = 0; r < 8; ++r)
                out[(size_t)(baseRow + r + (N_NODES_C - GARMENT_C)) * D_C + col]
                    = acc[t][r];
        }
    }
}

// -------------------------------------------------------------------------
// Kernel 2: weighted scatter-add over edges.
// One wave per edge: float4 gather of X0[dst] (512B/wave, fully coalesced),
// scale by edge_w, 4 non-returning agent-scope f32 atomic adds into out[src].
// Edge index forced scalar so edge-list loads become SMEM loads.
// -------------------------------------------------------------------------
__global__ __launch_bounds__(256) void pc_edge_scatter(const int*   __restrict__ esrc,
                                                       const int*   __restrict__ edst,
                                                       const float* __restrict__ ew,
                                                       const float* __restrict__ X0,
                                                       float* __restrict__ out,
                                                       int n_edges) {
    const int lane = threadIdx.x & 31;
    const int e = __builtin_amdgcn_readfirstlane(blockIdx.x * 8 + (threadIdx.x >> 5));
    if (e >= n_edges) return;

    const int   src = esrc[e];
    const int   dst = edst[e];
    const float a   = ew[e];

    const float4 x = ((const float4*)(X0 + (size_t)dst * D_C))[lane];

    float* p = out + (size_t)src * D_C + lane * 4;
    __hip_atomic_fetch_add(p + 0, a * x.x, __ATOMIC_RELAXED, __HIP_MEMORY_SCOPE_AGENT);
    __hip_atomic_fetch_add(p + 1, a * x.y, __ATOMIC_RELAXED, __HIP_MEMORY_SCOPE_AGENT);
    __hip_atomic_fetch_add(p + 2, a * x.z, __ATOMIC_RELAXED, __HIP_MEMORY_SCOPE_AGENT);
    __hip_atomic_fetch_add(p + 3, a * x.w, __ATOMIC_RELAXED, __HIP_MEMORY_SCOPE_AGENT);
}

// -------------------------------------------------------------------------
// Kernel 3: in-place ReLU over the first N_NODES rows of out.
// -------------------------------------------------------------------------
__global__ __launch_bounds__(256) void pc_relu(float* __restrict__ out, int n_vec4) {
    int i = blockIdx.x * 256 + threadIdx.x;
    if (i >= n_vec4) return;
    float4 v = ((float4*)out)[i];
    v.x = fmaxf(v.x, 0.0f);
    v.y = fmaxf(v.y, 0.0f);
    v.z = fmaxf(v.z, 0.0f);
    v.w = fmaxf(v.w, 0.0f);
    ((float4*)out)[i] = v;
}

// ---------------------------------------------------------------------------
// Launch: init(bias) -> gemm(wmma) -> edge(atomics) -> relu
// ---------------------------------------------------------------------------
extern "C" void kernel_launch(void* const* d_in, const int* in_sizes, int n_in,
                              void* d_out, int out_size, void* d_ws, size_t ws_size,
                              hipStream_t stream) {
    const float* notes = (const float*)d_in[0];
    const float* w     = (const float*)d_in[1];
    const float* b     = (const float*)d_in[2];
    const int*   esrc  = (const int*)  d_in[3];
    const int*   edst  = (const int*)  d_in[4];
    const float* ew    = (const float*)d_in[5];

    const int n_nodes = in_sizes[0] / D_C;           // 100000
    const int n_edges = in_sizes[3];                 // 1600000

    float* out = (float*)d_out;
    float* X0  = (float*)d_ws;                       // 100000*128 f32 = 51.2 MB

    // 0) accumulator rows [0, n_nodes) <- bias
    {
        const int nv4 = n_nodes * (D_C / 4);
        pc_init_bias<<<(nv4 + 255) / 256, 256, 0, stream>>>(b, out, nv4);
    }
    // 1) X0 = notes @ w (and concat tail of out)
    {
        const int tiles  = (n_nodes + 15) / 16;      // 6250
        const int blocks = (tiles + 7) / 8;          // 782
        pc_gemm_wmma<<<blocks, 256, 0, stream>>>(notes, w, X0, out, n_nodes);
    }
    // 2) scatter-add messages into accumulator
    {
        const int blocks = (n_edges + 7) / 8;        // 200000
        pc_edge_scatter<<<blocks, 256, 0, stream>>>(esrc, edst, ew, X0, out, n_edges);
    }
    // 3) ReLU in place on the accumulator portion
    {
        const int nv4 = n_nodes * (D_C / 4);
        pc_relu<<<(nv4 + 255) / 256, 256, 0, stream>>>(out, nv4);
    }
}